// PyGGIN_42322607735316
// MI455X (gfx1250) — compile-verified
//
#include <hip/hip_runtime.h>
#include <hip/hip_bf16.h>

typedef float v2f __attribute__((ext_vector_type(2)));
typedef float v8f __attribute__((ext_vector_type(8)));

#define N_NODES 100000
#define N_EDGES 400000
#define N_GRAPHS 4000
#define FEAT 75
#define DIM 128
#define H_TASK 512
#define LAYERS 5

// ---------------------------------------------------------------------------
// Pack B[K,N] (row-major) into WMMA fragment order, zero-padded to Kpad.
// Layout: Bp[((g*N + n)*2 + j)*2 + i] = B[4g + 2j + i][n]   (kp = 4g+2j+i)
// so a lane (half=j, col n) reads its v2f fragment with one b64 load and a
// constant stride of N*4 floats per k-step.
// ---------------------------------------------------------------------------
__global__ __launch_bounds__(256) void pack_b_kernel(
    const float* __restrict__ B, float* __restrict__ Bp, int K, int N)
{
    const int Kpad = (K + 3) & ~3;
    int id = blockIdx.x * 256 + threadIdx.x;
    if (id >= Kpad * N) return;
    int kp = id / N;
    int n  = id - kp * N;
    int g = kp >> 2, j = (kp >> 1) & 1, i = kp & 1;
    float v = (kp < K) ? B[(size_t)kp * N + n] : 0.0f;
    Bp[(((size_t)g * N + n) * 2 + j) * 2 + i] = v;
}

// ---------------------------------------------------------------------------
// GEMM: out[M,N] = act(A[M,K] @ B[K,N] + bias[N]) with B pre-packed (Bp).
// Block = 256 threads = 8 waves; block tile 32 rows x 128 cols (two 16-row
// WMMA subtiles sharing each B fragment -> 2 accumulators per wave).
// A staged in LDS in chunks of KC<=128 columns (zero-padded tail).
// grid.x = M/32, grid.y = N/128. Dynamic LDS = 32*KC*4 bytes.
// ---------------------------------------------------------------------------
__global__ __launch_bounds__(256) void gemm_f32_wmma_kernel(
    const float* __restrict__ A, const float* __restrict__ Bp,
    const float* __restrict__ bias, float* __restrict__ out,
    int M, int K, int N, int relu)
{
    extern __shared__ float ldsA[];   // 32 * KC floats
    const int tid  = threadIdx.x;
    const int wave = tid >> 5;
    const int lane = tid & 31;
    const int half = lane >> 4;       // 0 = lanes 0-15, 1 = lanes 16-31
    const int l16  = lane & 15;
    const int row0 = blockIdx.x * 32;
    const int n0   = blockIdx.y * 128 + wave * 16;
    const int Kpad = (K + 3) & ~3;
    const int KC   = (Kpad <= 128) ? Kpad : 128;

    // Accumulators initialized with bias (bias depends on output column only).
    const float bv = bias[n0 + l16];
    v8f c0, c1;
#pragma unroll
    for (int r = 0; r < 8; ++r) { c0[r] = bv; c1[r] = bv; }

    // Packed-B pointer for this lane; advances by N v2f units per k-step.
    const v2f* __restrict__ bp =
        (const v2f*)Bp + ((size_t)(n0 + l16) * 2 + half);

    for (int kc = 0; kc < Kpad; kc += KC) {
        __syncthreads();
        // Stage A chunk [32 x KC] into LDS (zero-pad columns >= K).
        for (int i = tid; i < 32 * KC; i += 256) {
            int r = i / KC, ccol = i - r * KC;
            int gc = kc + ccol;
            ldsA[i] = (gc < K) ? A[(size_t)(row0 + r) * K + gc] : 0.0f;
        }
        __syncthreads();

        int aoff = l16 * KC + half * 2;
        const int a1off = 16 * KC;
        const int steps = KC >> 2;
        for (int g = 0; g < steps; ++g) {
            v2f b = bp[0];
            v2f a0, a1;
            a0[0] = ldsA[aoff];         a0[1] = ldsA[aoff + 1];
            a1[0] = ldsA[aoff + a1off]; a1[1] = ldsA[aoff + a1off + 1];
            c0 = __builtin_amdgcn_wmma_f32_16x16x4_f32(
                    false, a0, false, b, (short)0, c0, false, false);
            c1 = __builtin_amdgcn_wmma_f32_16x16x4_f32(
                    false, a1, false, b, (short)0, c1, false, false);
            aoff += 4;
            bp += (size_t)N;           // next k-group, same column
        }
    }

    // C/D layout: VGPR r -> row r (lanes 0-15) or row r+8 (lanes 16-31).
#pragma unroll
    for (int r = 0; r < 8; ++r) {
        float v0 = c0[r], v1 = c1[r];
        if (relu) { v0 = fmaxf(v0, 0.0f); v1 = fmaxf(v1, 0.0f); }
        size_t col = n0 + l16;
        out[(size_t)(row0 + r + half * 8) * N + col]      = v0;
        out[(size_t)(row0 + 16 + r + half * 8) * N + col] = v1;
    }
}

// z0[dst] += h[src] over edges. 32 lanes per edge x 4 floats each (DIM=128).
__global__ __launch_bounds__(256) void scatter_add_kernel(
    const float* __restrict__ h, const int* __restrict__ src,
    const int* __restrict__ dst, float* __restrict__ agg, int E)
{
    int id = blockIdx.x * 256 + threadIdx.x;
    int e = id >> 5;
    if (e >= E) return;
    int c = (id & 31) * 4;
    const float* hp = h + (size_t)src[e] * DIM + c;
    float* ap = agg + (size_t)dst[e] * DIM + c;
#pragma unroll
    for (int i = 0; i < 4; ++i) atomicAdd(ap + i, hp[i]);
}

__global__ __launch_bounds__(256) void copy4_kernel(
    const float4* __restrict__ in, float4* __restrict__ out, int n4)
{
    int id = blockIdx.x * 256 + threadIdx.x;
    if (id < n4) out[id] = in[id];
}

__global__ __launch_bounds__(256) void zero_kernel(float* __restrict__ p, int n)
{
    int id = blockIdx.x * 256 + threadIdx.x;
    if (id < n) p[id] = 0.0f;
}

// g[batch[n]] += h[n] (global add pool), one thread per (node, feature).
__global__ __launch_bounds__(256) void pool_kernel(
    const float* __restrict__ h, const int* __restrict__ batch,
    float* __restrict__ g, int n_nodes)
{
    int id = blockIdx.x * 256 + threadIdx.x;
    int n = id >> 7;
    if (n >= n_nodes) return;
    int d = id & (DIM - 1);
    atomicAdd(&g[(size_t)batch[n] * DIM + d], h[id]);
}

// out[i] = dot(g2[i,:], w[:,0]) + b  (final 128 -> 1 layer)
__global__ __launch_bounds__(128) void final_dot_kernel(
    const float* __restrict__ g2, const float* __restrict__ w,
    const float* __restrict__ b, float* __restrict__ out, int G)
{
    int i = blockIdx.x * 128 + threadIdx.x;
    if (i >= G) return;
    float s = b[0];
    const float* row = g2 + (size_t)i * DIM;
#pragma unroll 4
    for (int k = 0; k < DIM; ++k) s = fmaf(row[k], w[k], s);
    out[i] = s;
}

static inline int ceil4(int k) { return (k + 3) & ~3; }

extern "C" void kernel_launch(void* const* d_in, const int* in_sizes, int n_in,
                              void* d_out, int out_size, void* d_ws, size_t ws_size,
                              hipStream_t stream) {
    const float* x       = (const float*)d_in[0];
    const int*   edge    = (const int*)d_in[1];   // [2, E]
    const int*   batch   = (const int*)d_in[2];
    const float* W_embed = (const float*)d_in[3];
    const float* b_embed = (const float*)d_in[4];
    const float* gin_W1  = (const float*)d_in[5];
    const float* gin_b1  = (const float*)d_in[6];
    const float* gin_W2  = (const float*)d_in[7];
    const float* gin_b2  = (const float*)d_in[8];
    const float* tW0     = (const float*)d_in[9];
    const float* tb0     = (const float*)d_in[10];
    const float* tW1     = (const float*)d_in[11];
    const float* tb1     = (const float*)d_in[12];
    const float* tW2     = (const float*)d_in[13];
    const float* tb2     = (const float*)d_in[14];
    float* out = (float*)d_out;

    const int* src = edge;
    const int* dst = edge + N_EDGES;

    // ---- Workspace layout -------------------------------------------------
    size_t nodeBuf = (size_t)N_NODES * DIM;
    float* h  = (float*)d_ws;             // current node features (51.2 MB)
    float* z0 = h + nodeBuf;              // h + agg                (51.2 MB)
    float* t  = z0 + nodeBuf;             // MLP intermediate       (51.2 MB)
    // Packed weights after the node buffers (~1.2 MB total).
    float* wp = t + nodeBuf;
    float* Wp_embed = wp;                              wp += (size_t)ceil4(FEAT) * DIM;
    float* Wp1      = wp;                              wp += (size_t)LAYERS * DIM * DIM;
    float* Wp2      = wp;                              wp += (size_t)LAYERS * DIM * DIM;
    float* tWp0     = wp;                              wp += (size_t)DIM * H_TASK;
    float* tWp1     = wp;                              wp += (size_t)H_TASK * DIM;
    // Graph-level buffers reuse z0/t space after the GIN layers.
    float* g  = z0;                                    // [G, 128]
    float* g1 = t;                                     // [G, 512]
    float* g2 = z0 + (size_t)N_GRAPHS * H_TASK;        // [G, 128]

    dim3 blk(256);

    // ---- Pack all weight matrices into WMMA fragment order ----------------
    {
        int n = ceil4(FEAT) * DIM;
        pack_b_kernel<<<(n + 255) / 256, blk, 0, stream>>>(W_embed, Wp_embed, FEAT, DIM);
        n = DIM * DIM;
        for (int l = 0; l < LAYERS; ++l) {
            pack_b_kernel<<<(n + 255) / 256, blk, 0, stream>>>(
                gin_W1 + (size_t)l * DIM * DIM, Wp1 + (size_t)l * DIM * DIM, DIM, DIM);
            pack_b_kernel<<<(n + 255) / 256, blk, 0, stream>>>(
                gin_W2 + (size_t)l * DIM * DIM, Wp2 + (size_t)l * DIM * DIM, DIM, DIM);
        }
        n = DIM * H_TASK;
        pack_b_kernel<<<(n + 255) / 256, blk, 0, stream>>>(tW0, tWp0, DIM, H_TASK);
        pack_b_kernel<<<(n + 255) / 256, blk, 0, stream>>>(tW1, tWp1, H_TASK, DIM);
    }

    // ---- 1) Node embedding: h = x @ W_embed + b_embed ---------------------
    {
        int KC = ceil4(FEAT);             // 76 (single chunk)
        dim3 grid(N_NODES / 32, DIM / 128);
        gemm_f32_wmma_kernel<<<grid, blk, 32 * KC * sizeof(float), stream>>>(
            x, Wp_embed, b_embed, h, N_NODES, FEAT, DIM, 0);
    }

    // ---- 2) GIN layers ----------------------------------------------------
    for (int l = 0; l < LAYERS; ++l) {
        int n4 = (int)(nodeBuf / 4);
        copy4_kernel<<<(n4 + 255) / 256, blk, 0, stream>>>(
            (const float4*)h, (float4*)z0, n4);          // z0 = (1+eps)*h, eps=0
        scatter_add_kernel<<<(N_EDGES * 32 + 255) / 256, blk, 0, stream>>>(
            h, src, dst, z0, N_EDGES);

        dim3 grid(N_NODES / 32, DIM / 128);
        gemm_f32_wmma_kernel<<<grid, blk, 32 * DIM * sizeof(float), stream>>>(
            z0, Wp1 + (size_t)l * DIM * DIM, gin_b1 + (size_t)l * DIM,
            t, N_NODES, DIM, DIM, 1);
        gemm_f32_wmma_kernel<<<grid, blk, 32 * DIM * sizeof(float), stream>>>(
            t, Wp2 + (size_t)l * DIM * DIM, gin_b2 + (size_t)l * DIM,
            h, N_NODES, DIM, DIM, (l < LAYERS - 1) ? 1 : 0);
    }

    // ---- 3) Global add pool ----------------------------------------------
    {
        int n = N_GRAPHS * DIM;
        zero_kernel<<<(n + 255) / 256, blk, 0, stream>>>(g, n);
        pool_kernel<<<(N_NODES * DIM + 255) / 256, blk, 0, stream>>>(
            h, batch, g, N_NODES);
    }

    // ---- 4) Task MLP ------------------------------------------------------
    {
        dim3 grid0(N_GRAPHS / 32, H_TASK / 128);
        gemm_f32_wmma_kernel<<<grid0, blk, 32 * DIM * sizeof(float), stream>>>(
            g, tWp0, tb0, g1, N_GRAPHS, DIM, H_TASK, 1);
        dim3 grid1(N_GRAPHS / 32, DIM / 128);
        gemm_f32_wmma_kernel<<<grid1, blk, 32 * 128 * sizeof(float), stream>>>(
            g1, tWp1, tb1, g2, N_GRAPHS, H_TASK, DIM, 1);
        final_dot_kernel<<<(N_GRAPHS + 127) / 128, dim3(128), 0, stream>>>(
            g2, tW2, tb2, out, N_GRAPHS);
    }
}